// Deform_Conv_36636071035629
// MI455X (gfx1250) — compile-verified
//
#include <hip/hip_runtime.h>

typedef __attribute__((ext_vector_type(2))) float v2f;
typedef __attribute__((ext_vector_type(8))) float v8f;

#define BB  8
#define CC  256
#define HH  64
#define WWd 64
#define OO  256
#define K2  9
#define HWX (HH * WWd)                 // 4096
#define NSL (K2 * CC / 4)              // 576 K-slices (4-wide) for main GEMM
#define APACK_FLOATS  (NSL * 16 * 64)  // 589824 : wd fragments
#define W1PACK_FLOATS (64 * 4 * 64)    // 16384  : w1 fragments
#define W2PACK_FLOATS (16 * 2 * 64)    // 2048   : w2 fragments (rows>=18 zero)
#define NPIX 32                        // pixels per workgroup (2 N-tiles)
#define SSTR 4128                      // skewed LDS region stride per N-tile

__device__ __forceinline__ int iclamp(int v, int lo, int hi) {
    return v < lo ? lo : (v > hi ? hi : v);
}

// ---------------------------------------------------------------------------
// Repack weights into WMMA 16x4 f32 A-fragment layout:
//   lane 0..15  : M = lane,      holds K+0 (v=0) and K+1 (v=1)
//   lane 16..31 : M = lane-16,   holds K+2 (v=0) and K+3 (v=1)
// ---------------------------------------------------------------------------
__global__ void repack_kernel(const float* __restrict__ w1,
                              const float* __restrict__ w2,
                              const float* __restrict__ wd,
                              float* __restrict__ ws) {
    int i = blockIdx.x * blockDim.x + threadIdx.x;
    if (i < APACK_FLOATS) {
        int v = i & 1, lane = (i >> 1) & 31, mt = (i >> 6) & 15, s = i >> 10;
        int k = s >> 6, cs = s & 63;                    // s = k*64 + cs
        int c = cs * 4 + ((lane >> 4) << 1) + v;
        int o = mt * 16 + (lane & 15);
        ws[i] = wd[(o * CC + c) * K2 + k];
    } else if (i < APACK_FLOATS + W1PACK_FLOATS) {
        int j = i - APACK_FLOATS;
        int v = j & 1, lane = (j >> 1) & 31, mt = (j >> 6) & 3, s = j >> 8;
        int c = s * 4 + ((lane >> 4) << 1) + v;
        int row = mt * 16 + (lane & 15);
        ws[i] = w1[row * CC + c];
    } else if (i < APACK_FLOATS + W1PACK_FLOATS + W2PACK_FLOATS) {
        int j = i - APACK_FLOATS - W1PACK_FLOATS;
        int v = j & 1, lane = (j >> 1) & 31, mt = (j >> 6) & 1, s = j >> 7;
        int hj = s * 4 + ((lane >> 4) << 1) + v;
        int row = mt * 16 + (lane & 15);
        ws[i] = (row < 18) ? w2[row * 64 + hj] : 0.0f;
    }
}

// ---------------------------------------------------------------------------
// Fused deformable conv. One workgroup = 32 consecutive pixels (same b,h),
// all 256 output channels. 4 waves; wave w owns M-tiles 4w..4w+3, both N-tiles.
// ---------------------------------------------------------------------------
__global__ __launch_bounds__(128) void deform_conv_kernel(
    const float* __restrict__ x,  const float* __restrict__ b1,
    const float* __restrict__ b2, const float* __restrict__ bd,
    const float* __restrict__ ws, float* __restrict__ out) {

    __shared__ float ldsS[2 * SSTR];    // sampled tile (2 skewed regions): 33 KB
    __shared__ float ldsH[64 * NPIX];   // hidden features h:               8 KB
    __shared__ float ldsOff[18 * NPIX]; // offsets dy/dx:                   2.3 KB

    const int tid    = threadIdx.x;
    const int wave   = tid >> 5;
    const int lane   = tid & 31;
    const int pixL   = lane & 15;
    const int hiHalf = lane >> 4;       // 0: K+0/K+1 half, 1: K+2/K+3 half

    const int t    = blockIdx.x;        // 0..1023
    const int b    = t >> 7;
    const int rem  = t & 127;
    const int hrow = rem >> 1;
    const int wseg = rem & 1;
    const int p0   = hrow * WWd + wseg * NPIX;

    const float* xb    = x + b * CC * HWX;
    const v2f*   apack = (const v2f*)ws;
    const v2f*   w1p   = (const v2f*)(ws + APACK_FLOATS);
    const v2f*   w2p   = (const v2f*)(ws + APACK_FLOATS + W1PACK_FLOATS);

    // ---- Phase 0a: h = relu(x . w1^T + b1)  (16 hidden rows/wave, 2 N-tiles)
    {
        v8f hC0 = {}, hC1 = {};
        const int c0l = hiHalf << 1;
        for (int s = 0; s < 64; ++s) {
            v2f a = w1p[(s * 4 + wave) * 32 + lane];
            int c = s * 4 + c0l;
            v2f bv0, bv1;
            bv0.x = xb[c * HWX + p0 + pixL];
            bv0.y = xb[(c + 1) * HWX + p0 + pixL];
            bv1.x = xb[c * HWX + p0 + 16 + pixL];
            bv1.y = xb[(c + 1) * HWX + p0 + 16 + pixL];
            hC0 = __builtin_amdgcn_wmma_f32_16x16x4_f32(false, a, false, bv0,
                                                        (short)0, hC0, false, false);
            hC1 = __builtin_amdgcn_wmma_f32_16x16x4_f32(false, a, false, bv1,
                                                        (short)0, hC1, false, false);
        }
#pragma unroll
        for (int r = 0; r < 8; ++r) {
            int jrow = wave * 16 + r + (hiHalf << 3);
            float bias = b1[jrow];
            ldsH[jrow * NPIX + pixL]      = fmaxf(hC0[r] + bias, 0.0f);
            ldsH[jrow * NPIX + 16 + pixL] = fmaxf(hC1[r] + bias, 0.0f);
        }
    }
    __syncthreads();

    // ---- Phase 0b: off = h . w2^T + b2  (M=18 pad 32; wave -> (mt, ntile))
    {
        const int mt  = wave & 1;
        const int ntB = wave >> 1;
        v8f oC = {};
        const int j0l = hiHalf << 1;
        for (int s = 0; s < 16; ++s) {
            v2f a = w2p[(s * 2 + mt) * 32 + lane];
            int j = s * 4 + j0l;
            v2f bv;
            bv.x = ldsH[j * NPIX + ntB * 16 + pixL];
            bv.y = ldsH[(j + 1) * NPIX + ntB * 16 + pixL];
            oC = __builtin_amdgcn_wmma_f32_16x16x4_f32(false, a, false, bv,
                                                       (short)0, oC, false, false);
        }
#pragma unroll
        for (int r = 0; r < 8; ++r) {
            int row = mt * 16 + r + (hiHalf << 3);
            if (row < 18)
                ldsOff[row * NPIX + ntB * 16 + pixL] = oC[r] + b2[row];
        }
    }
    __syncthreads();

    // ---- Phase 1: for each tap k: bilinear-sample 256x32 tile, WMMA-accum
    v8f acc[4][2] = {{v8f{}, v8f{}}, {v8f{}, v8f{}},
                     {v8f{}, v8f{}}, {v8f{}, v8f{}}};
    const int pixS = lane;              // 0..31 : pixel within tile
    const int clS  = wave;              // 0..3  : 64 channels per thread
    const int wwp  = (wseg << 5) + pixS;
    const int sBase = (pixS >> 4) * SSTR;   // skewed region for this pixel
    const int pS16  = pixS & 15;

    for (int k = 0; k < K2; ++k) {
        // -- cooperative bilinear sampling into LDS (conflict-free skew)
        {
            float dy = ldsOff[(2 * k) * NPIX + pixS];
            float dx = ldsOff[(2 * k + 1) * NPIX + pixS];
            int  ky  = k / 3, kx = k - ky * 3;
            float yf = (float)(hrow + ky - 1) + dy;
            float xf = (float)(wwp + kx - 1) + dx;
            float y0f = floorf(yf), x0f = floorf(xf);
            float wy = yf - y0f, wx = xf - x0f;
            int y0 = (int)y0f, x0 = (int)x0f;
            int y1 = y0 + 1,   x1 = x0 + 1;
            float vy0 = (y0 >= 0 && y0 < HH)  ? 1.0f : 0.0f;
            float vy1 = (y1 >= 0 && y1 < HH)  ? 1.0f : 0.0f;
            float vx0 = (x0 >= 0 && x0 < WWd) ? 1.0f : 0.0f;
            float vx1 = (x1 >= 0 && x1 < WWd) ? 1.0f : 0.0f;
            float w00 = (1.f - wy) * (1.f - wx) * vy0 * vx0;
            float w01 = (1.f - wy) * wx * vy0 * vx1;
            float w10 = wy * (1.f - wx) * vy1 * vx0;
            float w11 = wy * wx * vy1 * vx1;
            int y0c = iclamp(y0, 0, HH - 1),  y1c = iclamp(y1, 0, HH - 1);
            int x0c = iclamp(x0, 0, WWd - 1), x1c = iclamp(x1, 0, WWd - 1);
            int a00 = y0c * WWd + x0c, a01 = y0c * WWd + x1c;
            int a10 = y1c * WWd + x0c, a11 = y1c * WWd + x1c;
#pragma unroll 4
            for (int j = 0; j < 64; ++j) {
                int c = clS * 64 + j;
                const float* xc = xb + c * HWX;
                float v = w00 * xc[a00] + w01 * xc[a01] +
                          w10 * xc[a10] + w11 * xc[a11];
                ldsS[sBase + c * 16 + pS16] = v;
            }
        }
        __syncthreads();

        // -- WMMA accumulate over 64 channel-slices of 4 (8 WMMA / 4 A-loads)
        {
            const int c0l = hiHalf << 1;
            const v2f* ap = apack + (size_t)(k * 64) * 16 * 32;
            for (int cs = 0; cs < 64; ++cs) {
                int cb = cs * 4 + c0l;
                v2f bv0, bv1;
                bv0.x = ldsS[cb * 16 + pixL];
                bv0.y = ldsS[(cb + 1) * 16 + pixL];
                bv1.x = ldsS[SSTR + cb * 16 + pixL];
                bv1.y = ldsS[SSTR + (cb + 1) * 16 + pixL];
                const v2f* aps = ap + cs * 16 * 32;
#pragma unroll
                for (int m = 0; m < 4; ++m) {
                    v2f a = aps[(wave * 4 + m) * 32 + lane];
                    acc[m][0] = __builtin_amdgcn_wmma_f32_16x16x4_f32(
                        false, a, false, bv0, (short)0, acc[m][0], false, false);
                    acc[m][1] = __builtin_amdgcn_wmma_f32_16x16x4_f32(
                        false, a, false, bv1, (short)0, acc[m][1], false, false);
                }
            }
        }
        __syncthreads();
    }

    // ---- Epilogue: add bias, coalesced store (both N-tiles)
#pragma unroll
    for (int m = 0; m < 4; ++m) {
#pragma unroll
        for (int r = 0; r < 8; ++r) {
            int o = (wave * 4 + m) * 16 + r + (hiHalf << 3);
            float bias = bd[o];
            int base = (b * OO + o) * HWX + p0;
            out[base + pixL]      = acc[m][0][r] + bias;
            out[base + 16 + pixL] = acc[m][1][r] + bias;
        }
    }
}

extern "C" void kernel_launch(void* const* d_in, const int* in_sizes, int n_in,
                              void* d_out, int out_size, void* d_ws, size_t ws_size,
                              hipStream_t stream) {
    const float* x  = (const float*)d_in[0];
    const float* w1 = (const float*)d_in[1];
    const float* b1 = (const float*)d_in[2];
    const float* w2 = (const float*)d_in[3];
    const float* b2 = (const float*)d_in[4];
    const float* wd = (const float*)d_in[5];
    const float* bd = (const float*)d_in[6];
    float* out = (float*)d_out;
    float* ws  = (float*)d_ws;

    const int total = APACK_FLOATS + W1PACK_FLOATS + W2PACK_FLOATS;
    repack_kernel<<<(total + 255) / 256, 256, 0, stream>>>(w1, w2, wd, ws);
    deform_conv_kernel<<<BB * HWX / NPIX, 128, 0, stream>>>(x, b1, b2, bd, ws, out);
}